// eppConstrainer_background_26139170964129
// MI455X (gfx1250) — compile-verified
//
#include <hip/hip_runtime.h>
#include <hip/hip_bf16.h>

#define BZ 16
#define H  384
#define W  1024
#define HW (H * W)                         // 393216
#define BPB 32                             // blocks per batch
#define NB  (BZ * BPB)                     // 512 blocks
#define TPB 256                            // 8 waves (wave32)
#define GROUPS_PER_BATCH (HW / 4)          // 98304 float4-groups
#define GROUPS_PER_BLOCK (GROUPS_PER_BATCH / BPB)  // 3072 -> 12 iters/thread

typedef __attribute__((ext_vector_type(2))) float v2f;
typedef __attribute__((ext_vector_type(8))) float v8f;

// ---------------------------------------------------------------------------
// Wave-level f32 sum using V_WMMA_F32_16X16X4_F32.
// A: a.x = per-lane value -> A[M][0] (lanes 0-15), A[M][2] (lanes 16-31);
//    a.y = 0              -> A[M][1], A[M][3] = 0.
// B: all ones (layout-independent). D[M][N] = v[M] + v[M+16].
// D VGPR r: lanes 0-15 = T[r], lanes 16-31 = T[r+8]; sum the 8 VGPRs, then
// one xor-16 shuffle folds the two halves -> every lane holds the wave sum.
// Requires EXEC all-ones: callers have uniform control flow.
// ---------------------------------------------------------------------------
__device__ __forceinline__ float wave_sum_wmma(float v) {
    v2f a; a.x = v;    a.y = 0.0f;
    v2f b; b.x = 1.0f; b.y = 1.0f;
    v8f c = {};
    v8f d = __builtin_amdgcn_wmma_f32_16x16x4_f32(
        /*neg_a=*/false, a, /*neg_b=*/false, b,
        /*c_mod=*/(short)0, c, /*reuse_a=*/false, /*reuse_b=*/false);
    float s = d[0] + d[1] + d[2] + d[3] + d[4] + d[5] + d[6] + d[7];
    s += __shfl_xor(s, 16, 32);
    return s;
}

// ---------------------------------------------------------------------------
// Kernel 1: build E_b = Kinv^T @ (skew(t/|t|) @ R) @ Kinv  (16 batches)
// ---------------------------------------------------------------------------
__global__ void epp_prep_E_kernel(const float* __restrict__ Kall,
                                  const float* __restrict__ Pall,
                                  float* __restrict__ Eout) {
    int bidx = threadIdx.x;
    if (bidx >= BZ) return;
    const float* k = Kall + bidx * 9;
    const float* p = Pall + bidx * 16;

    float R00 = p[0],  R01 = p[1],  R02 = p[2];
    float R10 = p[4],  R11 = p[5],  R12 = p[6];
    float R20 = p[8],  R21 = p[9],  R22 = p[10];
    float t0 = p[3], t1 = p[7], t2 = p[11];
    float inv_n = rsqrtf(t0 * t0 + t1 * t1 + t2 * t2);
    t0 *= inv_n; t1 *= inv_n; t2 *= inv_n;

    // F = skew(t) @ R ; skew(t) = [[0,-t2,t1],[t2,0,-t0],[-t1,t0,0]]
    float F00 = -t2 * R10 + t1 * R20, F01 = -t2 * R11 + t1 * R21, F02 = -t2 * R12 + t1 * R22;
    float F10 =  t2 * R00 - t0 * R20, F11 =  t2 * R01 - t0 * R21, F12 =  t2 * R02 - t0 * R22;
    float F20 = -t1 * R00 + t0 * R10, F21 = -t1 * R01 + t0 * R11, F22 = -t1 * R02 + t0 * R12;

    // Kinv via adjugate
    float a = k[0], b = k[1], c = k[2];
    float d = k[3], e = k[4], f = k[5];
    float g = k[6], h = k[7], i = k[8];
    float det = a * (e * i - f * h) - b * (d * i - f * g) + c * (d * h - e * g);
    float id = 1.0f / det;
    float KI00 = (e * i - f * h) * id, KI01 = (c * h - b * i) * id, KI02 = (b * f - c * e) * id;
    float KI10 = (f * g - d * i) * id, KI11 = (a * i - c * g) * id, KI12 = (c * d - a * f) * id;
    float KI20 = (d * h - e * g) * id, KI21 = (b * g - a * h) * id, KI22 = (a * e - b * d) * id;

    // M = F @ Kinv
    float M00 = F00 * KI00 + F01 * KI10 + F02 * KI20;
    float M01 = F00 * KI01 + F01 * KI11 + F02 * KI21;
    float M02 = F00 * KI02 + F01 * KI12 + F02 * KI22;
    float M10 = F10 * KI00 + F11 * KI10 + F12 * KI20;
    float M11 = F10 * KI01 + F11 * KI11 + F12 * KI21;
    float M12 = F10 * KI02 + F11 * KI12 + F12 * KI22;
    float M20 = F20 * KI00 + F21 * KI10 + F22 * KI20;
    float M21 = F20 * KI01 + F21 * KI11 + F22 * KI21;
    float M22 = F20 * KI02 + F21 * KI12 + F22 * KI22;

    // E = Kinv^T @ M : E[r][cC] = sum_k KI[k][r] * M[k][cC]
    float* E = Eout + bidx * 9;
    E[0] = KI00 * M00 + KI10 * M10 + KI20 * M20;
    E[1] = KI00 * M01 + KI10 * M11 + KI20 * M21;
    E[2] = KI00 * M02 + KI10 * M12 + KI20 * M22;
    E[3] = KI01 * M00 + KI11 * M10 + KI21 * M20;
    E[4] = KI01 * M01 + KI11 * M11 + KI21 * M21;
    E[5] = KI01 * M02 + KI11 * M12 + KI21 * M22;
    E[6] = KI02 * M00 + KI12 * M10 + KI22 * M20;
    E[7] = KI02 * M01 + KI12 * M11 + KI22 * M21;
    E[8] = KI02 * M02 + KI12 * M12 + KI22 * M22;
}

// ---------------------------------------------------------------------------
// Kernel 2: streaming quadratic-form + masked reduction (HBM-bound, b128 loads)
// ---------------------------------------------------------------------------
__global__ void __launch_bounds__(TPB)
epp_main_kernel(const float* __restrict__ flow,
                const float* __restrict__ valid,
                const float* __restrict__ Emat,
                float* __restrict__ partials) {
    const int bidx = blockIdx.x / BPB;   // batch
    const int blk  = blockIdx.x % BPB;   // sub-block within batch

    const float* Eb = Emat + bidx * 9;   // broadcast load, L2-resident
    const float E00 = Eb[0], E01 = Eb[1], E02 = Eb[2];
    const float E10 = Eb[3], E11 = Eb[4], E12 = Eb[5];
    const float E20 = Eb[6], E21 = Eb[7], E22 = Eb[8];

    const float4* fx4 = reinterpret_cast<const float4*>(flow + (size_t)bidx * 2 * HW);
    const float4* fy4 = fx4 + (HW / 4);
    const float4* vv4 = reinterpret_cast<const float4*>(valid + (size_t)bidx * HW);

    const int gend = (blk + 1) * GROUPS_PER_BLOCK;
    float sA = 0.0f, sV = 0.0f;

    // Uniform trip count (3072 groups / 256 threads = 12) -> no divergence,
    // EXEC is all-ones at the WMMA reduction below.
    for (int gg = blk * GROUPS_PER_BLOCK + threadIdx.x; gg < gend; gg += TPB) {
        float4 fx = fx4[gg];            // global_load_b128 x3, fully coalesced
        float4 fy = fy4[gg];
        float4 vv = vv4[gg];
        int rem = gg << 2;              // first pixel of the 4-group
        float yf = (float)(rem >> 10);  // W = 1024
        float xf = (float)(rem & (W - 1));
        // hoist the y-dependent part of E*pts
        float c0 = fmaf(E01, yf, E02);
        float c1 = fmaf(E11, yf, E12);
        float c2 = fmaf(E21, yf, E22);

#define EPP_PIX(K, FXC, FYC, VC)                                         \
        {                                                                \
            float x  = xf + (float)(K);                                  \
            float r0 = fmaf(E00, x, c0);                                 \
            float r1 = fmaf(E10, x, c1);                                 \
            float r2 = fmaf(E20, x, c2);                                 \
            float ev = fmaf(x + (FXC), r0, fmaf(yf + (FYC), r1, r2));    \
            sA = fmaf(fabsf(ev), (VC), sA);                              \
            sV += (VC);                                                  \
        }
        EPP_PIX(0, fx.x, fy.x, vv.x)
        EPP_PIX(1, fx.y, fy.y, vv.y)
        EPP_PIX(2, fx.z, fy.z, vv.z)
        EPP_PIX(3, fx.w, fy.w, vv.w)
#undef EPP_PIX
    }

    // wave32 reduction via v_wmma_f32_16x16x4_f32 (exact f32)
    float wA = wave_sum_wmma(sA);
    float wV = wave_sum_wmma(sV);

    __shared__ float shA[TPB / 32];
    __shared__ float shV[TPB / 32];
    const int wave = threadIdx.x >> 5;
    const int lane = threadIdx.x & 31;
    if (lane == 0) { shA[wave] = wA; shV[wave] = wV; }
    __syncthreads();
    if (threadIdx.x == 0) {
        float bA = 0.0f, bV = 0.0f;
#pragma unroll
        for (int wv = 0; wv < TPB / 32; ++wv) { bA += shA[wv]; bV += shV[wv]; }
        partials[2 * blockIdx.x]     = bA;
        partials[2 * blockIdx.x + 1] = bV;
    }
}

// ---------------------------------------------------------------------------
// Kernel 3: deterministic fixed-tree finalize of NB=512 partial pairs
// ---------------------------------------------------------------------------
__global__ void __launch_bounds__(TPB)
epp_finalize_kernel(const float* __restrict__ partials, float* __restrict__ out) {
    __shared__ float sA[TPB];
    __shared__ float sV[TPB];
    const int t = threadIdx.x;
    // NB == 2*TPB: fold two pairs per thread
    float a = partials[2 * t]     + partials[2 * (t + TPB)];
    float v = partials[2 * t + 1] + partials[2 * (t + TPB) + 1];
    sA[t] = a; sV[t] = v;
    __syncthreads();
    for (int s = TPB / 2; s > 0; s >>= 1) {
        if (t < s) { sA[t] += sA[t + s]; sV[t] += sV[t + s]; }
        __syncthreads();
    }
    if (t == 0) out[0] = sA[0] / (sV[0] + 1.0f);
}

// ---------------------------------------------------------------------------
extern "C" void kernel_launch(void* const* d_in, const int* in_sizes, int n_in,
                              void* d_out, int out_size, void* d_ws, size_t ws_size,
                              hipStream_t stream) {
    (void)in_sizes; (void)n_in; (void)out_size; (void)ws_size;
    const float* flow  = (const float*)d_in[0];  // (16, 2, 384, 1024)
    const float* intr  = (const float*)d_in[1];  // (16, 3, 3)
    const float* pose  = (const float*)d_in[2];  // (16, 4, 4)
    const float* valid = (const float*)d_in[3];  // (16, 384, 1024)

    float* ws       = (float*)d_ws;
    float* Emat     = ws;        // 144 floats used
    float* partials = ws + 256;  // 2*NB = 1024 floats

    epp_prep_E_kernel<<<1, 32, 0, stream>>>(intr, pose, Emat);
    epp_main_kernel<<<NB, TPB, 0, stream>>>(flow, valid, Emat, partials);
    epp_finalize_kernel<<<1, TPB, 0, stream>>>(partials, (float*)d_out);
}